// HGCN_6133213299293
// MI455X (gfx1250) — compile-verified
//
#include <hip/hip_runtime.h>
#include <cstdint>

typedef float v2f __attribute__((ext_vector_type(2)));
typedef float v4f __attribute__((ext_vector_type(4)));
typedef float v8f __attribute__((ext_vector_type(8)));

// Guaranteed hardware float atomic (no CAS fallback): GLOBAL_ATOMIC_ADD_F32, no return.
__device__ __forceinline__ void atomic_add_f32_hw(float* p, float v) {
    asm volatile("global_atomic_add_f32 %0, %1, off" :: "v"(p), "v"(v) : "memory");
}

// ---------------------------------------------------------------------------
// GEMM: out[M, NT*16] = preop(A)[M,128] @ W[128, NT*16] + bias
// preop (layer 2): a -> max(0, -|curv| * a)   (fused relu+scale of agg1)
//
// W is staged into LDS *K-pair interleaved*: W[k][c] lives at
// ldsW[((k>>1)*NCOL + c)*2 + (k&1)], so a B fragment {W[k+kb][c], W[k+kb+1][c]}
// is one aligned ds_load_b64 straight into an even VGPR pair (no re-pairing movs).
// The A stream is register double-buffered at distance 2 K-steps so each
// global_load_b64 is covered by ~16 WMMAs + 16 DS loads.
// ---------------------------------------------------------------------------
template <int NT, bool PRE_RELU>
__global__ __launch_bounds__(256) void hgcn_gemm_wmma(
    const float* __restrict__ A, const float* __restrict__ W,
    const float* __restrict__ bias, const float* __restrict__ curv,
    float* __restrict__ out, int M) {
    constexpr int K = 128;
    constexpr int NCOL = NT * 16;
    __shared__ float ldsW[K * NCOL];  // K-pair interleaved

    for (int i = threadIdx.x; i < K * NCOL; i += 256) {
        const int k = i / NCOL, c = i - k * NCOL;
        ldsW[(((k >> 1) * NCOL + c) << 1) + (k & 1)] = W[i];
    }
    __syncthreads();

    const int lane = threadIdx.x & 31;
    const int wave = threadIdx.x >> 5;
    const int tile = blockIdx.x * 8 + wave;
    if (tile * 16 >= M) return;       // uniform per wave: EXEC stays all-1s

    const int mn = lane & 15;         // A row / B,C col within tile
    const int kh = lane >> 4;         // K-half selector: 0 or 1
    float cs = 0.0f;
    if constexpr (PRE_RELU) cs = -fabsf(curv[0]);

    v8f acc[NT];
#pragma unroll
    for (int t = 0; t < NT; ++t) acc[t] = (v8f)0.0f;

    // lane's A fragment base: row (tile*16 + mn), K sub-offset 2*kh
    const float* __restrict__ arow =
        A + (size_t)(tile * 16 + mn) * K + (kh << 1);

    auto lda = [&](int k) -> v2f {
        v2f a = *(const v2f*)(arow + k);       // global_load_b64
        if constexpr (PRE_RELU) {
            a.x = fmaxf(cs * a.x, 0.0f);
            a.y = fmaxf(cs * a.y, 0.0f);
        }
        return a;
    };
    auto wmma_group = [&](v2f a, int k) {
        const int kp = (k >> 1) + kh;          // interleaved K-pair index
#pragma unroll
        for (int t = 0; t < NT; ++t) {
            v2f b = *(const v2f*)((const float*)ldsW +
                                  ((kp * NCOL + t * 16 + mn) << 1));  // ds_load_b64
            acc[t] = __builtin_amdgcn_wmma_f32_16x16x4_f32(
                false, a, false, b, (short)0, acc[t], false, false);
        }
    };

    // software pipeline, prefetch distance 2 K-steps
    v2f a0 = lda(0);
    v2f a1 = lda(4);
    int k = 0;
    for (; k < K - 8; k += 4) {
        v2f a2 = lda(k + 8);
        wmma_group(a0, k);
        a0 = a1;
        a1 = a2;
    }
    wmma_group(a0, k);        // k = K-8
    wmma_group(a1, k + 4);    // k = K-4

    // C/D layout: VGPR r, lane L -> row = 8*(L>>4) + r, col = L&15
    const int mhi = kh * 8;
#pragma unroll
    for (int t = 0; t < NT; ++t) {
        const float bc = bias[t * 16 + mn];
        float* __restrict__ orow =
            out + (size_t)(tile * 16 + mhi) * NCOL + t * 16 + mn;
#pragma unroll
        for (int r = 0; r < 8; ++r) orow[(size_t)r * NCOL] = acc[t][r] + bc;
    }
}

// ---------------------------------------------------------------------------
// Edge scatter: agg[dst[e]] += H[src[e]]  (one wave per edge; self loops are
// edge ids >= E). NCOL = 32*VEC; vectorized b128/b64 gather (L2-resident) +
// hardware f32 atomics.
// ---------------------------------------------------------------------------
template <int VEC>
__global__ __launch_bounds__(256) void hgcn_scatter_add(
    const float* __restrict__ H, const int* __restrict__ src,
    const int* __restrict__ dst, float* __restrict__ agg, int E, int N) {
    constexpr int NCOL = 32 * VEC;
    const int w = (int)((blockIdx.x * 256u + threadIdx.x) >> 5);
    const int lane = threadIdx.x & 31;
    if (w >= E + N) return;
    int s, d;
    if (w < E) { s = src[w]; d = dst[w]; }
    else       { s = w - E; d = s; }          // appended self loop
    const float* __restrict__ hp = H + (size_t)s * NCOL + lane * VEC;
    float* __restrict__ ap = agg + (size_t)d * NCOL + lane * VEC;
    if constexpr (VEC == 4) {
        v4f v = *(const v4f*)hp;
        atomic_add_f32_hw(ap + 0, v.x);
        atomic_add_f32_hw(ap + 1, v.y);
        atomic_add_f32_hw(ap + 2, v.z);
        atomic_add_f32_hw(ap + 3, v.w);
    } else {
        v2f v = *(const v2f*)hp;
        atomic_add_f32_hw(ap + 0, v.x);
        atomic_add_f32_hw(ap + 1, v.y);
    }
}

// ---------------------------------------------------------------------------
// out = log_softmax(-|curv| * agg, axis=1), 64 cols: one wave per row,
// 2 elements per lane, wave32 shuffle reductions.
// ---------------------------------------------------------------------------
__global__ __launch_bounds__(256) void hgcn_logsoftmax64(
    const float* __restrict__ agg, const float* __restrict__ curv,
    float* __restrict__ out, int N) {
    const int w = (int)((blockIdx.x * 256u + threadIdx.x) >> 5);
    const int lane = threadIdx.x & 31;
    if (w >= N) return;
    const float c = -fabsf(curv[0]);
    const float* __restrict__ rp = agg + (size_t)w * 64;
    float v0 = c * rp[lane];
    float v1 = c * rp[lane + 32];
    float m = fmaxf(v0, v1);
#pragma unroll
    for (int off = 16; off > 0; off >>= 1) m = fmaxf(m, __shfl_xor(m, off, 32));
    float s = __expf(v0 - m) + __expf(v1 - m);
#pragma unroll
    for (int off = 16; off > 0; off >>= 1) s += __shfl_xor(s, off, 32);
    const float lse = m + __logf(s);
    float* __restrict__ op = out + (size_t)w * 64;
    op[lane] = v0 - lse;
    op[lane + 32] = v1 - lse;
}

// ---------------------------------------------------------------------------
extern "C" void kernel_launch(void* const* d_in, const int* in_sizes, int n_in,
                              void* d_out, int out_size, void* d_ws, size_t ws_size,
                              hipStream_t stream) {
    const float* x     = (const float*)d_in[0];
    const int*   ei    = (const int*)  d_in[1];
    const float* W1    = (const float*)d_in[2];
    const float* b1    = (const float*)d_in[3];
    // d_in[4] (Wa1), d_in[5] (ba1), d_in[9] (Wa2), d_in[10] (ba2): dead math
    // (softmax over a size-1 axis is identically 1.0) -> intentionally unused.
    const float* curv1 = (const float*)d_in[6];
    const float* W2    = (const float*)d_in[7];
    const float* b2    = (const float*)d_in[8];
    const float* curv2 = (const float*)d_in[11];

    const int N = in_sizes[0] / 128;   // 50000 nodes
    const int E = in_sizes[1] / 2;     // 800000 edges
    const int* src = ei;
    const int* dst = ei + E;

    char* ws = (char*)d_ws;
    float* H1   = (float*)ws; ws += (size_t)N * 128 * sizeof(float);
    float* AGG1 = (float*)ws; ws += (size_t)N * 128 * sizeof(float);
    float* H2   = (float*)ws; ws += (size_t)N * 64 * sizeof(float);
    float* AGG2 = (float*)ws;

    hipMemsetAsync(AGG1, 0, (size_t)N * 128 * sizeof(float), stream);
    hipMemsetAsync(AGG2, 0, (size_t)N * 64 * sizeof(float), stream);

    const int tiles      = (N + 15) / 16;        // 3125 (N % 16 == 0)
    const int gemmBlocks = (tiles + 7) / 8;
    const int scatBlocks = ((E + N) + 7) / 8;
    const int lsBlocks   = (N + 7) / 8;

    // Layer 1: H1 = X @ W1 + b1
    hgcn_gemm_wmma<8, false><<<gemmBlocks, 256, 0, stream>>>(x, W1, b1, nullptr, H1, N);
    // agg1 = scatter-add of H1[src] at dst (+ self loops)
    hgcn_scatter_add<4><<<scatBlocks, 256, 0, stream>>>(H1, src, dst, AGG1, E, N);
    // Layer 2: H2 = relu(-|c1| * agg1) @ W2 + b2   (relu+scale fused into A-load)
    hgcn_gemm_wmma<4, true><<<gemmBlocks, 256, 0, stream>>>(AGG1, W2, b2, curv1, H2, N);
    hgcn_scatter_add<2><<<scatBlocks, 256, 0, stream>>>(H2, src, dst, AGG2, E, N);
    // out = log_softmax(-|c2| * agg2)
    hgcn_logsoftmax64<<<lsBlocks, 256, 0, stream>>>(AGG2, curv2, (float*)d_out, N);
}